// GaussianSplat2D_55619826483253
// MI455X (gfx1250) — compile-verified
//
#include <hip/hip_runtime.h>
#include <hip/hip_bf16.h>
#include <math.h>

typedef __attribute__((ext_vector_type(2))) float v2f;
typedef __attribute__((ext_vector_type(4))) float v4f;
typedef __attribute__((ext_vector_type(8))) float v8f;

#define N_GAUSS   1024
#define IMG_H     256
#define IMG_W     256
#define EPS_COV   1e-6f
#define ALPHA_MAX 0.999f

#define GF_STRIDE 10                       // padded g(n) row stride in dwords (bank spread, 8B aligned pairs)
#define GF_DW     (N_GAUSS * GF_STRIDE)    // 10240 dwords
#define CP_DW     (N_GAUSS * 4)            // 4096 dwords (r,g,b,opacity)
#define SMEM_DW   (GF_DW + CP_DW)          // 14336 dwords = 56 KB LDS

// ---------------------------------------------------------------------------
// Kernel 1: per-gaussian preprocessing.
// Builds the rank-6 feature vector g(n) such that power(p,n) = f(p) . g(n)
// with f(p) = [x^2, x, y^2, y, x*y, 1, 0, 0], plus packed color/opacity.
// ---------------------------------------------------------------------------
__global__ __launch_bounds__(256) void gs_prep(const float* __restrict__ means,
                                               const float* __restrict__ log_scales,
                                               const float* __restrict__ rot,
                                               const float* __restrict__ colors,
                                               const float* __restrict__ opac,
                                               float* __restrict__ ws) {
  int n = blockIdx.x * blockDim.x + threadIdx.x;
  if (n >= N_GAUSS) return;

  float mx = means[2 * n + 0];
  float my = means[2 * n + 1];
  float s2x = __expf(2.f * log_scales[2 * n + 0]);   // exp(ls)^2
  float s2y = __expf(2.f * log_scales[2 * n + 1]);
  float s, c;
  __sincosf(rot[n], &s, &c);

  // cov = R diag(s2) R^T + eps*I,  R = [[c,-s],[s,c]]
  float a = c * c * s2x + s * s * s2y + EPS_COV;
  float b = c * s * (s2x - s2y);
  float d = s * s * s2x + c * c * s2y + EPS_COV;
  float inv = 1.f / (a * d - b * b);
  float ia = d * inv, ib = -b * inv, id = a * inv;

  float* gf = ws + (size_t)n * GF_STRIDE;
  gf[0] = -0.5f * ia;                     // * x^2
  gf[1] = ia * mx + ib * my;              // * x
  gf[2] = -0.5f * id;                     // * y^2
  gf[3] = id * my + ib * mx;              // * y
  gf[4] = -ib;                            // * x*y
  gf[5] = -0.5f * (ia * mx * mx + 2.f * ib * mx * my + id * my * my);  // * 1
  gf[6] = 0.f; gf[7] = 0.f;               // K padding (6 -> 8)
  gf[8] = 0.f; gf[9] = 0.f;               // bank-spread padding

  float* cp = ws + GF_DW + (size_t)n * 4;
  cp[0] = colors[3 * n + 0];
  cp[1] = colors[3 * n + 1];
  cp[2] = colors[3 * n + 2];
  cp[3] = opac[n];
}

// ---------------------------------------------------------------------------
// Kernel 2: render. 256 threads = 8 waves; each wave owns 16 pixels.
// Per 16-gaussian chunk: two v_wmma_f32_16x16x4_f32 build the 16x16 power
// tile (A rows = gaussians, B cols = pixels). D layout: lane L holds pixel
// col = L&15, gaussian rows = VGPR + 8*(L>>4). Sequential alpha compositing
// runs locally over 8 gaussians per lane, merged across lane halves with one
// shfl_xor(16) per chunk.
// ---------------------------------------------------------------------------
__global__ __launch_bounds__(256) void gs_render(const float* __restrict__ ws,
                                                 float* __restrict__ out) {
  __shared__ float smem[SMEM_DW];

  // Cooperative float4 copy of all gaussian params into LDS.
  {
    const v4f* src = (const v4f*)ws;
    v4f* dst = (v4f*)smem;
    for (int i = threadIdx.x; i < SMEM_DW / 4; i += 256) dst[i] = src[i];
  }
  __syncthreads();

  const float* sg = smem;            // [1024][GF_STRIDE] feature rows
  const float* sc = smem + GF_DW;    // [1024][4] r,g,b,opacity

  const int lane = threadIdx.x & 31;
  const int wave = threadIdx.x >> 5;
  const int half = lane >> 4;        // 0: K 0..1 / rows 0..7; 1: K 2..3 / rows 8..15
  const int col  = lane & 15;

  const int p = blockIdx.x * 128 + wave * 16 + col;   // pixel (lane pair shares p)
  const float x = (float)(p & (IMG_W - 1));
  const float y = (float)(p >> 8);

  // B-matrix fragments f(p): loop-invariant.
  const float f[8] = { x * x, x, y * y, y, x * y, 1.f, 0.f, 0.f };
  const v2f bA = { f[2 * half + 0], f[2 * half + 1] };         // K = 0..3
  const v2f bB = { f[4 + 2 * half + 0], f[4 + 2 * half + 1] }; // K = 4..7

  float T = 1.f;                 // running transmittance for pixel p
  float cr = 0.f, cg = 0.f, cb = 0.f;

  for (int gb = 0; gb < N_GAUSS; gb += 16) {
    // A-matrix fragments: gaussian row = gb + col (16x4 f32 A layout:
    // lanes 0-15 -> K0,K1; lanes 16-31 -> K2,K3).
    const float* gf = sg + (gb + col) * GF_STRIDE;
    const v2f aA = { gf[2 * half + 0], gf[2 * half + 1] };
    const v2f aB = { gf[4 + 2 * half + 0], gf[4 + 2 * half + 1] };

    v8f P = {};
    P = __builtin_amdgcn_wmma_f32_16x16x4_f32(false, aA, false, bA,
                                              (short)0, P, false, false);
    P = __builtin_amdgcn_wmma_f32_16x16x4_f32(false, aB, false, bB,
                                              (short)0, P, false, false);

    const int gg = gb + 8 * half;   // first gaussian row held by this lane
    float w[8];
    float q = 1.f;                  // local prod(1 - alpha)
#pragma unroll
    for (int j = 0; j < 8; ++j) {
      float op = sc[(gg + j) * 4 + 3];
      float al = fminf(op * __expf(P[j]), ALPHA_MAX);
      w[j] = al * q;
      q *= (1.f - al);
    }

    // Merge the two lane halves: upper half (gaussians 8..15) starts after
    // the lower half's tail product.
    float qo  = __shfl_xor(q, 16, 32);
    float pre = half ? (T * qo) : T;
    T *= q * qo;

#pragma unroll
    for (int j = 0; j < 8; ++j) {
      float wv = w[j] * pre;
      const float* cc = sc + (gg + j) * 4;
      cr = fmaf(wv, cc[0], cr);
      cg = fmaf(wv, cc[1], cg);
      cb = fmaf(wv, cc[2], cb);
    }
  }

  // Each pixel's color is split across the lane pair; reduce and store.
  cr += __shfl_xor(cr, 16, 32);
  cg += __shfl_xor(cg, 16, 32);
  cb += __shfl_xor(cb, 16, 32);
  if (half == 0) {
    float* o = out + (size_t)p * 3;
    o[0] = cr; o[1] = cg; o[2] = cb;
  }
}

// ---------------------------------------------------------------------------
extern "C" void kernel_launch(void* const* d_in, const int* in_sizes, int n_in,
                              void* d_out, int out_size, void* d_ws, size_t ws_size,
                              hipStream_t stream) {
  const float* means      = (const float*)d_in[0];
  const float* log_scales = (const float*)d_in[1];
  const float* rotations  = (const float*)d_in[2];
  const float* colors     = (const float*)d_in[3];
  const float* opacities  = (const float*)d_in[4];
  float* ws = (float*)d_ws;

  gs_prep<<<N_GAUSS / 256, 256, 0, stream>>>(means, log_scales, rotations,
                                             colors, opacities, ws);
  gs_render<<<(IMG_H * IMG_W) / 128, 256, 0, stream>>>(ws, (float*)d_out);
}